// CostVolume_80762565034123
// MI455X (gfx1250) — compile-verified
//
#include <hip/hip_runtime.h>

typedef float v2f __attribute__((ext_vector_type(2)));
typedef float v8f __attribute__((ext_vector_type(8)));

#define B_ 8
#define C_ 128
#define H_ 96
#define W_ 320
#define DISP 48
#define CH_STRIDE (H_ * W_)      // 30720 floats between channels in (b,c,h,w)
#define OUT_MSTRIDE (H_ * W_ + 1)  // out-offset step per m along a band tile row

// One wave computes one 16x16 band tile of G = L @ R^T for one (b,h) slice
// using V_WMMA_F32_16X16X4_F32 accumulation over c=128 (32 K-steps).
// Tiles whose x' origin is negative are exactly the zero triangle (x < i).
__global__ __launch_bounds__(128) void cost_volume_wmma(
    const float* __restrict__ L, const float* __restrict__ R,
    float* __restrict__ out) {
  const int lane = threadIdx.x & 31;
  const int cb   = threadIdx.x >> 5;  // column block 0..3 (one per wave)
  const int hf   = lane >> 4;         // lane half: 0 or 1
  const int col  = lane & 15;         // n (and spatial offset for frag loads)

  const int rb = blockIdx.x % 20;     // row block (x tile)
  const int bh = blockIdx.x / 20;
  const int h  = bh % H_;
  const int b  = bh / H_;

  const int x0    = rb * 16;               // x tile origin (m rows)
  const int xp0   = x0 - 48 + cb * 16;     // x' tile origin (n cols)
  const int idiag = 48 - 16 * cb;          // i = idiag + m - n

  // Output offset: o(m) = obase + m * OUT_MSTRIDE  (valid only when i in range)
  const int obase = ((b * DISP + idiag - col) * H_ + h) * W_ + x0;
  const int mbase = 8 * hf;

  if (xp0 < 0) {
    // Entire tile lies in x' < 0 region -> cost is zero where i is valid.
#pragma unroll
    for (int r = 0; r < 8; ++r) {
      int m = r + mbase;
      int i = idiag + m - col;
      if ((unsigned)i < DISP) out[obase + m * OUT_MSTRIDE] = 0.0f;
    }
    return;
  }

  // Per-lane fragment pointers: A holds L[x0+col, k+2*hf+{0,1}],
  // B holds R[xp0+col, k+2*hf+{0,1}] (A/B lane layouts coincide for f32 K=4).
  const size_t plane = (size_t)b * C_ * CH_STRIDE + (size_t)h * W_;
  const float* lp = L + plane + (size_t)(2 * hf) * CH_STRIDE + (x0 + col);
  const float* rp = R + plane + (size_t)(2 * hf) * CH_STRIDE + (xp0 + col);

  v8f acc = {};
#pragma unroll 8
  for (int k = 0; k < C_; k += 4) {
    v2f a, bb;
    a[0]  = lp[0];
    a[1]  = lp[CH_STRIDE];
    bb[0] = rp[0];
    bb[1] = rp[CH_STRIDE];
    // 8 args: (neg_a, A, neg_b, B, c_mod, C, reuse_a, reuse_b)
    acc = __builtin_amdgcn_wmma_f32_16x16x4_f32(false, a, false, bb,
                                                (short)0, acc, false, false);
    lp += 4 * CH_STRIDE;
    rp += 4 * CH_STRIDE;
  }

  // D layout: lane l, VGPR r -> m = r + 8*(l>>4), n = l&15.
  const float scale = 1.0f / (float)C_;
#pragma unroll
  for (int r = 0; r < 8; ++r) {
    int m = r + mbase;
    int i = idiag + m - col;
    if ((unsigned)i < DISP) out[obase + m * OUT_MSTRIDE] = acc[r] * scale;
  }
}

extern "C" void kernel_launch(void* const* d_in, const int* in_sizes, int n_in,
                              void* d_out, int out_size, void* d_ws,
                              size_t ws_size, hipStream_t stream) {
  const float* left  = (const float*)d_in[0];
  const float* right = (const float*)d_in[1];
  float* out = (float*)d_out;

  dim3 grid(B_ * H_ * 20);  // 15360 blocks: (b,h) slices x 20 row blocks
  dim3 block(128);          // 4 waves: one per band column block
  hipLaunchKernelGGL(cost_volume_wmma, grid, block, 0, stream, left, right,
                     out);
}